// SPGAT_6751688589922
// MI455X (gfx1250) — compile-verified
//
#include <hip/hip_runtime.h>
#include <float.h>

typedef __attribute__((ext_vector_type(2))) float v2f;
typedef __attribute__((ext_vector_type(8))) float v8f;

#define F_IN  256
#define F_OUT 64
#define WT_STRIDE (F_IN + 4)   // 260: pads LDS rows so 16 lanes spread over all 64 banks
#define LRELU_ALPHA 0.2f
#define GAT_EPS 1e-15f

// float atomic max via signed-max / unsigned-min trick (correct for mixed signs)
__device__ __forceinline__ void atomicMaxF32(float* addr, float val) {
    if (val >= 0.0f) {
        atomicMax((int*)addr, __float_as_int(val));
    } else {
        atomicMin((unsigned int*)addr, (unsigned int)__float_as_int(val));
    }
}

// ---------------------------------------------------------------------------
// Kernel 0: init accumulators (harness poisons ws with 0xAA; we must zero)
// ---------------------------------------------------------------------------
__global__ __launch_bounds__(256) void gat_init(float4* __restrict__ hprime4,
                                                float* __restrict__ rowsum,
                                                float* __restrict__ gmax,
                                                long long nh4, int N) {
    const long long i = (long long)blockIdx.x * 256 + threadIdx.x;
    if (i < nh4) hprime4[i] = make_float4(0.f, 0.f, 0.f, 0.f);
    if (i < N)   rowsum[i] = 0.0f;
    if (i == 0)  gmax[0] = -FLT_MAX;
}

// ---------------------------------------------------------------------------
// Kernel 1: h = X @ W via V_WMMA_F32_16X16X4_F32, fused s1 = h.a1, s2 = h.a2
// One wave -> one 16(M) x 64(N) tile; K in steps of 4; branch-free K loop
// (tail rows clamped, never stored). W transposed+padded in LDS so every
// B fragment is a single contiguous ds_load_b64 into an aligned VGPR pair.
// ---------------------------------------------------------------------------
__global__ __launch_bounds__(256) void gat_gemm_wmma(
    const float* __restrict__ X, const float* __restrict__ W,
    const float* __restrict__ av, float* __restrict__ h,
    float* __restrict__ s1, float* __restrict__ s2, int N)
{
    __shared__ float WT[F_OUT * WT_STRIDE];   // WT[n][k] = W[k][n], ~66.6 KB

    for (int i = threadIdx.x; i < F_IN * F_OUT; i += 256) {
        const int k = i >> 6;      // / F_OUT
        const int n = i & (F_OUT - 1);
        WT[n * WT_STRIDE + k] = W[i];
    }
    __syncthreads();

    const int wave = threadIdx.x >> 5;
    const int lane = threadIdx.x & 31;
    const int half = lane >> 4;        // 0: lanes 0-15 (K+0,1), 1: lanes 16-31 (K+2,3)
    const int l16  = lane & 15;
    const int row0 = blockIdx.x * 128 + wave * 16;
    if (row0 >= N) return;            // wave-uniform

    const int m = row0 + l16;
    const float* __restrict__ xrow =
        X + (size_t)(m < N ? m : N - 1) * F_IN + half * 2;   // clamped: no divergence
    const float* __restrict__ wt0 = &WT[l16 * WT_STRIDE + half * 2];

    v8f acc[4] = {};
#pragma unroll 4
    for (int k0 = 0; k0 < F_IN; k0 += 4) {
        const v2f af = *(const v2f*)(xrow + k0);             // global_load_b64
#pragma unroll
        for (int t = 0; t < 4; ++t) {
            const v2f bf = *(const v2f*)(wt0 + t * 16 * WT_STRIDE + k0); // ds_load_b64
            acc[t] = __builtin_amdgcn_wmma_f32_16x16x4_f32(
                false, af, false, bf, (short)0, acc[t], false, false);
        }
    }

    // ---- fused attention projections: s1[m] = sum_n h[m][n]*a1[n], s2 likewise
    float a1v[4], a2v[4];
#pragma unroll
    for (int t = 0; t < 4; ++t) {
        a1v[t] = av[t * 16 + l16];
        a2v[t] = av[F_OUT + t * 16 + l16];
    }
    float p[8], q[8];
#pragma unroll
    for (int s = 0; s < 8; ++s) {
        p[s] = acc[0][s] * a1v[0] + acc[1][s] * a1v[1] +
               acc[2][s] * a1v[2] + acc[3][s] * a1v[3];
        q[s] = acc[0][s] * a2v[0] + acc[1][s] * a2v[1] +
               acc[2][s] * a2v[2] + acc[3][s] * a2v[3];
    }
#pragma unroll
    for (int off = 1; off < 16; off <<= 1) {   // xor masks < 16 stay within halves
#pragma unroll
        for (int s = 0; s < 8; ++s) {
            p[s] += __shfl_xor(p[s], off);
            q[s] += __shfl_xor(q[s], off);
        }
    }

    // ---- store h tile (C layout: lane = col, slot = row, halves split M 0-7/8-15)
#pragma unroll
    for (int t = 0; t < 4; ++t) {
#pragma unroll
        for (int s = 0; s < 8; ++s) {
            const int mm = row0 + half * 8 + s;
            if (mm < N) h[(size_t)mm * F_OUT + t * 16 + l16] = acc[t][s];
        }
    }
    if (l16 == 0) {
#pragma unroll
        for (int s = 0; s < 8; ++s) {
            const int mm = row0 + half * 8 + s;
            if (mm < N) { s1[mm] = p[s]; s2[mm] = q[s]; }
        }
    }
}

// ---------------------------------------------------------------------------
// Kernel 2: per-edge leaky-relu logits + global max (1 atomic per block)
// ---------------------------------------------------------------------------
__global__ __launch_bounds__(256) void gat_edge_logits(
    const long long* __restrict__ src, const long long* __restrict__ dst,
    const float* __restrict__ s1, const float* __restrict__ s2,
    float* __restrict__ ev, float* __restrict__ gmax, long long E)
{
    __shared__ float wmax[8];
    const long long e = (long long)blockIdx.x * 256 + threadIdx.x;
    float v = -FLT_MAX;
    if (e < E) {
        const float x = s1[src[e]] + s2[dst[e]];
        v = (x > 0.0f) ? x : LRELU_ALPHA * x;
        ev[e] = v;
    }
#pragma unroll
    for (int off = 16; off > 0; off >>= 1)
        v = fmaxf(v, __shfl_xor(v, off));
    if ((threadIdx.x & 31) == 0) wmax[threadIdx.x >> 5] = v;
    __syncthreads();
    if (threadIdx.x == 0) {
        float m = wmax[0];
#pragma unroll
        for (int i = 1; i < 8; ++i) m = fmaxf(m, wmax[i]);
        atomicMaxF32(gmax, m);
    }
}

// ---------------------------------------------------------------------------
// Kernel 3: edge aggregation. 16 lanes / edge, float4 gather of h[dst],
// native f32 atomic adds (global_atomic_add_f32) into L2-resident h_prime.
// ---------------------------------------------------------------------------
__global__ __launch_bounds__(256) void gat_aggregate(
    const long long* __restrict__ src, const long long* __restrict__ dst,
    const float* __restrict__ ev, const float* __restrict__ gmax,
    const float* __restrict__ h, float* __restrict__ hprime,
    float* __restrict__ rowsum, long long E)
{
    const int grp = threadIdx.x >> 4;   // 16 edges per block
    const int l   = threadIdx.x & 15;   // 4 features per lane
    const long long e = (long long)blockIdx.x * 16 + grp;
    if (e >= E) return;

    const long long si = src[e];
    const long long di = dst[e];
    const float w = expf(ev[e] - gmax[0]);

    const float4 x = ((const float4*)(h + (size_t)di * F_OUT))[l];
    float* out = hprime + (size_t)si * F_OUT + l * 4;
    unsafeAtomicAdd(out + 0, w * x.x);
    unsafeAtomicAdd(out + 1, w * x.y);
    unsafeAtomicAdd(out + 2, w * x.z);
    unsafeAtomicAdd(out + 3, w * x.w);
    if (l == 0) unsafeAtomicAdd(&rowsum[si], w);
}

// ---------------------------------------------------------------------------
// Kernel 4: finalize: divide by rowsum+eps, ELU (float4 vectorized)
// ---------------------------------------------------------------------------
__global__ __launch_bounds__(256) void gat_finalize(
    const float4* __restrict__ hprime4, const float* __restrict__ rowsum,
    float4* __restrict__ out4, long long nh4)
{
    const long long i = (long long)blockIdx.x * 256 + threadIdx.x;
    if (i >= nh4) return;
    const long long n = i >> 4;                 // 16 float4 per node
    const float inv = 1.0f / (rowsum[n] + GAT_EPS);
    float4 v = hprime4[i];
    v.x *= inv; v.y *= inv; v.z *= inv; v.w *= inv;
    v.x = (v.x > 0.0f) ? v.x : expm1f(v.x);
    v.y = (v.y > 0.0f) ? v.y : expm1f(v.y);
    v.z = (v.z > 0.0f) ? v.z : expm1f(v.z);
    v.w = (v.w > 0.0f) ? v.w : expm1f(v.w);
    out4[i] = v;
}

// ---------------------------------------------------------------------------
extern "C" void kernel_launch(void* const* d_in, const int* in_sizes, int n_in,
                              void* d_out, int out_size, void* d_ws, size_t ws_size,
                              hipStream_t stream) {
    const float*     X     = (const float*)d_in[0];
    const long long* edges = (const long long*)d_in[1];
    const float*     W     = (const float*)d_in[2];
    const float*     av    = (const float*)d_in[3];

    const int       N = in_sizes[0] / F_IN;
    const long long E = (long long)in_sizes[1] / 2;
    const long long* src = edges;        // edge_index[0]
    const long long* dst = edges + E;    // edge_index[1]

    // workspace layout (floats)
    float* ws = (float*)d_ws;
    size_t off = 0;
    float* h      = ws + off; off += (size_t)N * F_OUT;  // 25.6 MB
    float* s1     = ws + off; off += (size_t)N;
    float* s2     = ws + off; off += (size_t)N;
    float* ev     = ws + off; off += (size_t)E;          // 12.8 MB
    float* rowsum = ws + off; off += (size_t)N;
    float* hprime = ws + off; off += (size_t)N * F_OUT;  // 25.6 MB
    float* gmax   = ws + off; off += 1;

    const long long nh  = (long long)N * F_OUT;
    const long long nh4 = nh / 4;

    gat_init<<<(int)((nh4 + 255) / 256), 256, 0, stream>>>(
        (float4*)hprime, rowsum, gmax, nh4, N);

    gat_gemm_wmma<<<(N + 127) / 128, 256, 0, stream>>>(X, W, av, h, s1, s2, N);

    gat_edge_logits<<<(int)((E + 255) / 256), 256, 0, stream>>>(
        src, dst, s1, s2, ev, gmax, E);

    gat_aggregate<<<(int)((E + 15) / 16), 256, 0, stream>>>(
        src, dst, ev, gmax, h, hprime, rowsum, E);

    gat_finalize<<<(int)((nh4 + 255) / 256), 256, 0, stream>>>(
        (const float4*)hprime, rowsum, (float4*)d_out, nh4);
}